// LanczosInterpolationLayer_35673998361181
// MI455X (gfx1250) — compile-verified
//
#include <hip/hip_runtime.h>
#include <hip/hip_bf16.h>
#include <cmath>

typedef float v2f __attribute__((ext_vector_type(2)));
typedef float v8f __attribute__((ext_vector_type(8)));

#define IN_H 512
#define IN_W 512
#define OUT_H 1024
#define OUT_W 1024
#define NBC   96                 // 32 images * 3 channels
#define TILES_PER_BC 4096        // 64 x 64 tiles of 16x16
#define WAVES_PER_BLOCK 8
#define LDS_STRIDE 18            // even (8B-aligned pairs) and coprime-enough: 18*o mod 64 distinct for o=0..15

// Host-computed Lanczos4 tap tables, passed by value as kernel argument.
// w[0][*]: fractional offset f=0.75 (even output row/col), w[1][*]: f=0.25 (odd).
struct WTab { float w[2][8]; };

// ---------- monotone float<->uint encoding for integer atomic min/max ----------
__device__ __forceinline__ unsigned enc_f32(float f) {
    unsigned u = __float_as_uint(f);
    return (u & 0x80000000u) ? ~u : (u | 0x80000000u);
}
__device__ __forceinline__ float dec_f32(unsigned u) {
    return __uint_as_float((u & 0x80000000u) ? (u ^ 0x80000000u) : ~u);
}

// Build the shift-invariant 16x16 weight matrix W in LDS (one entry per thread
// of a 256-thread block), then have each wave pull its WMMA fragment.
// W[o][i] = tap (i - ibase(o)) of the parity-(o&1) vector, ibase(o) = ((o-1)>>1)+1.
__device__ __forceinline__ void build_wmat_lds(float* __restrict__ Wmat,
                                               const WTab wt, unsigned tid) {
    int o = (int)(tid >> 4);          // 0..15  (row of W)
    int i = (int)(tid & 15u);         // 0..15  (col of W)
    int t = i - (((o - 1) >> 1) + 1); // tap index
    float val = 0.0f;
    #pragma unroll
    for (int k = 0; k < 8; ++k) {
        float wv = (o & 1) ? wt.w[1][k] : wt.w[0][k];  // SGPR consts + cndmask
        val = (t == k) ? wv : val;
    }
    Wmat[o * LDS_STRIDE + i] = val;
}

// Per-lane WMMA fragment of W. Serves as A in pass 1 (W*X) and as B in pass 2
// (T*W^T): lane%16 indexes W's row, (chunk c, component j, lane-group g) give
// W's column K = 4c + j + 2g  (V_WMMA_F32_16X16X4_F32 operand layout).
__device__ __forceinline__ void load_wfrag(const float* __restrict__ Wmat,
                                           int lane, v2f wfrag[4]) {
    int o = lane & 15;
    int g = lane >> 4;
    #pragma unroll
    for (int c = 0; c < 4; ++c) {
        int idx = o * LDS_STRIDE + 4 * c + 2 * g;   // even -> 8B aligned pair
        wfrag[c].x = Wmat[idx];
        wfrag[c].y = Wmat[idx + 1];
    }
}

// Compute one 16x16 output tile of the separable Lanczos4 2x upsample.
// Result is in WMMA C/D layout: element (row = vr + 8*(lane>>4), col = lane&15)
// lives in component vr of the returned v8f.
__device__ __forceinline__ v8f resize_tile(const float* __restrict__ x,
                                           int bc, int toh, int tow,
                                           int lane, float* __restrict__ tile,
                                           const v2f wfrag[4]) {
    int g = lane >> 4;
    int j = lane & 15;
    const float* __restrict__ src = x + (size_t)bc * (IN_H * IN_W);

    int r0 = toh * 8 - 4;                          // input window origin (rows)
    int c0 = tow * 8 - 4;                          // input window origin (cols)
    int cc = min(max(c0 + j, 0), IN_W - 1);        // replicate border (cols)

    // ---- pass 1: T = W (16x16) * X_win (16x16), K split into 4 chunks of 4 ----
    v8f acc = {};
    #pragma unroll
    for (int c = 0; c < 4; ++c) {
        int k0 = 4 * c + 2 * g;
        int rA = min(max(r0 + k0,     0), IN_H - 1);  // replicate border (rows)
        int rB = min(max(r0 + k0 + 1, 0), IN_H - 1);
        v2f b;
        b.x = src[(size_t)rA * IN_W + cc];
        b.y = src[(size_t)rB * IN_W + cc];
        acc = __builtin_amdgcn_wmma_f32_16x16x4_f32(
                false, wfrag[c], false, b, (short)0, acc, false, false);
    }

    // ---- D-layout -> A-layout transpose through wave-private LDS tile ----
    #pragma unroll
    for (int vr = 0; vr < 8; ++vr) {
        int row = vr + 8 * g;
        tile[row * LDS_STRIDE + j] = acc[vr];
    }
    asm volatile("s_wait_dscnt 0" ::: "memory");   // wave-local LDS RAW fence

    // ---- pass 2: Out = T (16x16) * W^T (16x16); B operand == wfrag ----
    v8f acc2 = {};
    #pragma unroll
    for (int c = 0; c < 4; ++c) {
        int k0 = 4 * c + 2 * g;
        v2f a;
        a.x = tile[j * LDS_STRIDE + k0];
        a.y = tile[j * LDS_STRIDE + k0 + 1];
        acc2 = __builtin_amdgcn_wmma_f32_16x16x4_f32(
                false, a, false, wfrag[c], (short)0, acc2, false, false);
    }
    return acc2;
}

__device__ __forceinline__ void decode_wave(unsigned tid, unsigned bid,
                                            int& wave, int& lane,
                                            int& bc, int& toh, int& tow) {
    wave = (int)(tid >> 5);
    lane = (int)(tid & 31u);
    unsigned wg = bid * WAVES_PER_BLOCK + (unsigned)wave;
    bc  = (int)(wg >> 12);          // / 4096 ; uniform across the block (8 | 4096)
    unsigned rem = wg & 4095u;
    toh = (int)(rem >> 6);
    tow = (int)(rem & 63u);
}

// ---------- kernel 0: init per-image min/max workspace ----------
__global__ void lanczos_init_mm(unsigned* __restrict__ mm) {
    int i = threadIdx.x;
    if (i < 32) {
        mm[i * 2 + 0] = 0xFFFFFFFFu;   // identity for atomicMin
        mm[i * 2 + 1] = 0x00000000u;   // identity for atomicMax
    }
}

// ---------- kernel 1: resize + per-image min/max reduction ----------
__global__ void lanczos_resize_minmax(const float* __restrict__ x,
                                      unsigned* __restrict__ mm,
                                      const WTab wt) {
    __shared__ float Wmat[16 * LDS_STRIDE];
    __shared__ float tiles[WAVES_PER_BLOCK * 16 * LDS_STRIDE];
    __shared__ float red_mn[WAVES_PER_BLOCK], red_mx[WAVES_PER_BLOCK];

    int wave, lane, bc, toh, tow;
    decode_wave(threadIdx.x, blockIdx.x, wave, lane, bc, toh, tow);

    build_wmat_lds(Wmat, wt, threadIdx.x);
    __syncthreads();

    v2f wf[4];
    load_wfrag(Wmat, lane, wf);
    v8f v = resize_tile(x, bc, toh, tow, lane,
                        &tiles[wave * 16 * LDS_STRIDE], wf);

    float mn = v[0], mx = v[0];
    #pragma unroll
    for (int i = 1; i < 8; ++i) { mn = fminf(mn, v[i]); mx = fmaxf(mx, v[i]); }
    #pragma unroll
    for (int off = 16; off >= 1; off >>= 1) {
        mn = fminf(mn, __shfl_xor(mn, off, 32));
        mx = fmaxf(mx, __shfl_xor(mx, off, 32));
    }
    if (lane == 0) { red_mn[wave] = mn; red_mx[wave] = mx; }
    __syncthreads();
    if (threadIdx.x == 0) {
        float m = red_mn[0], M = red_mx[0];
        #pragma unroll
        for (int w = 1; w < WAVES_PER_BLOCK; ++w) {
            m = fminf(m, red_mn[w]);
            M = fmaxf(M, red_mx[w]);
        }
        int img = bc / 3;                    // uniform across block
        atomicMin(&mm[img * 2 + 0], enc_f32(m));
        atomicMax(&mm[img * 2 + 1], enc_f32(M));
    }
}

// ---------- kernel 2: recompute resize (L2-resident input), renormalize, emit ----------
__global__ void lanczos_finalize(const float* __restrict__ x,
                                 const unsigned* __restrict__ mm,
                                 float* __restrict__ out,
                                 const WTab wt) {
    __shared__ float Wmat[16 * LDS_STRIDE];
    __shared__ float tiles[WAVES_PER_BLOCK * 16 * LDS_STRIDE];

    int wave, lane, bc, toh, tow;
    decode_wave(threadIdx.x, blockIdx.x, wave, lane, bc, toh, tow);

    build_wmat_lds(Wmat, wt, threadIdx.x);
    __syncthreads();

    v2f wf[4];
    load_wfrag(Wmat, lane, wf);
    v8f v = resize_tile(x, bc, toh, tow, lane,
                        &tiles[wave * 16 * LDS_STRIDE], wf);

    int img = bc / 3;
    float mn = dec_f32(mm[img * 2 + 0]);
    float mx = dec_f32(mm[img * 2 + 1]);
    float range = mx - mn;
    const float ott = 1.0f / 255.0f;     // over_top_target_range
    const float ubt = -10.0f / 255.0f;   // under_bottom_target_range
    float otr = mx - 1.0f;
    float safe_otr = (otr != 0.0f) ? otr : 1.0f;
    float safe_ubr = (mn  != 0.0f) ? mn  : 1.0f;
    float otr_k = ott / safe_otr;
    float ubr_k = ubt / safe_ubr;
    bool renorm = (range > 1.0f);

    int g = lane >> 4;
    int p = lane & 15;
    size_t base_out = ((size_t)bc * OUT_H + (size_t)toh * 16) * OUT_W
                    + (size_t)tow * 16 + (size_t)p;

    #pragma unroll
    for (int vr = 0; vr < 8; ++vr) {
        float y = v[vr];
        float r = y;
        if (renorm) {
            float v1 = (y > 1.0f) ? ((y - 1.0f) * otr_k + (1.0f - ott))
                     : ((y < 1.0f) ? y * (1.0f - ott) : y);
            float v2 = (v1 < 0.0f) ? (v1 * ubr_k + ubt)
                     : ((v1 > 0.0f) ? v1 * (1.0f - ubt) : v1);
            r = v2;
        }
        r *= 255.0f;
        unsigned char u8 = (unsigned char)(int)r;   // numpy-style uint8 cast
        int row = vr + 8 * g;
        out[base_out + (size_t)row * OUT_W] = (float)u8;
    }
}

// ---------- host-side weight table (double precision, no device sinf) ----------
static WTab make_wtab() {
    WTab t;
    const double PI = 3.14159265358979323846;
    for (int par = 0; par < 2; ++par) {
        double f = par ? 0.25 : 0.75;    // parity of output index
        double w[8], s = 0.0;
        for (int k = 0; k < 8; ++k) {
            double d = f + 3.0 - (double)k;          // taps -3..4
            double s1 = (d == 0.0)       ? 1.0 : std::sin(PI * d) / (PI * d);
            double d4 = d * 0.25;
            double s2 = (d4 == 0.0)      ? 1.0 : std::sin(PI * d4) / (PI * d4);
            w[k] = s1 * s2;                          // |d| < 4 always
            s += w[k];
        }
        for (int k = 0; k < 8; ++k) t.w[par][k] = (float)(w[k] / s);
    }
    return t;
}

extern "C" void kernel_launch(void* const* d_in, const int* in_sizes, int n_in,
                              void* d_out, int out_size, void* d_ws, size_t ws_size,
                              hipStream_t stream) {
    (void)in_sizes; (void)n_in; (void)out_size; (void)ws_size;
    const float* x   = (const float*)d_in[0];
    float*       out = (float*)d_out;
    unsigned*    mm  = (unsigned*)d_ws;   // 32 images x {min,max} encoded u32

    const WTab wt = make_wtab();          // deterministic, host-side

    lanczos_init_mm<<<1, 32, 0, stream>>>(mm);

    // 96 image-channels * 4096 tiles, 8 waves (tiles) per 256-thread block
    const unsigned total_tiles = NBC * TILES_PER_BC;          // 393216
    const unsigned blocks = total_tiles / WAVES_PER_BLOCK;    // 49152
    lanczos_resize_minmax<<<dim3(blocks), dim3(256), 0, stream>>>(x, mm, wt);
    lanczos_finalize<<<dim3(blocks), dim3(256), 0, stream>>>(x, mm, out, wt);
}